// DynamicSphericalTorch_2542620640000
// MI455X (gfx1250) — compile-verified
//
#include <hip/hip_runtime.h>
#include <cstdint>

// Problem constants from the reference: B=64, N=16384, D=32, STEPS=6
#define N_NODES 16384
#define BATCH   64
#define DEG     32
#define NSTEPS  6
#define NODES_PER_BLOCK 4   // 4 nodes x 64-thread team = 256 threads/block
#define TILE_N  64          // init-kernel transpose tile (64 nodes x 64 batch)

typedef float v4f __attribute__((ext_vector_type(4)));

#if defined(__HIP_DEVICE_COMPILE__) && defined(__gfx1250__)
#define USE_TDM 1
#else
#define USE_TDM 0
#endif

#if USE_TDM
typedef unsigned int v4u __attribute__((ext_vector_type(4)));
typedef int          v8i __attribute__((ext_vector_type(8)));
typedef int          v4i __attribute__((ext_vector_type(4)));

// Tensor Data Mover: DMA a contiguous run of n_dwords DWORDs from global to LDS.
// D# per CDNA5 ISA ch.8: group0 = {count|flags, lds_addr, gaddr lo, gaddr hi|type=2},
// group1 = {data_size=4B, tensor_dim0=n, tensor_dim1=1, tile_dim0=n, tile_dim1=1, stride=n}.
__device__ __forceinline__ void tdm_load_row_to_lds(uint32_t lds_off,
                                                    const void* gptr,
                                                    uint32_t n_dwords)
{
  const uint64_t ga = (uint64_t)(uintptr_t)gptr;
  v4u g0;
  g0[0] = 1u;                                                   // count=1 valid user D#
  g0[1] = lds_off;                                              // LDS byte address
  g0[2] = (uint32_t)ga;                                         // global_addr[31:0]
  g0[3] = (uint32_t)((ga >> 32) & 0x1FFFFFFull) | (2u << 30);   // addr[56:32] | type=2
  v8i g1;
  g1[0] = (int)(2u << 16);          // data_size = 2 -> 4 bytes; no multicast/pad/iterate
  g1[1] = (int)(n_dwords << 16);    // tensor_dim0[15:0] (bits 63:48)
  g1[2] = (int)(1u << 16);          // tensor_dim1 = 1
  g1[3] = (int)(n_dwords << 16);    // tile_dim0 = n
  g1[4] = 1;                        // tile_dim1 = 1, tile_dim2 = 0
  g1[5] = (int)n_dwords;            // tensor_dim0_stride
  g1[6] = 0;
  g1[7] = 0;
  v4i g2 = {0, 0, 0, 0};
  v4i g3 = {0, 0, 0, 0};
#if __clang_major__ >= 23
  v8i g4 = {0, 0, 0, 0, 0, 0, 0, 0};
  __builtin_amdgcn_tensor_load_to_lds(g0, g1, g2, g3, g4, 0);
#else
  __builtin_amdgcn_tensor_load_to_lds(g0, g1, g2, g3, 0);
#endif
}
#endif // USE_TDM

// s0^T[n*64+b] = tanh(x[b,n]*w_in[n] + b0[n]) via LDS tile transpose:
// both the x read and the sT write are fully coalesced.
__global__ __launch_bounds__(256) void init_state_kernel(
    const float* __restrict__ x,
    const float* __restrict__ w_in,
    const float* __restrict__ b0,
    float* __restrict__ sT)
{
  __shared__ float tile[BATCH][TILE_N + 1];   // +1 pad: conflict-free transpose
  const uint32_t n0 = blockIdx.x * TILE_N;

#pragma unroll
  for (int it = 0; it < (BATCH * TILE_N) / 256; ++it) {
    const uint32_t i  = it * 256u + threadIdx.x;
    const uint32_t b  = i >> 6;          // batch row
    const uint32_t nn = i & 63u;         // node within tile (fastest -> coalesced)
    tile[b][nn] = x[b * N_NODES + n0 + nn];
  }
  __syncthreads();

#pragma unroll
  for (int it = 0; it < (BATCH * TILE_N) / 256; ++it) {
    const uint32_t j  = it * 256u + threadIdx.x;
    const uint32_t nn = j >> 6;
    const uint32_t b  = j & 63u;         // batch fastest -> coalesced sT store
    const uint32_t n  = n0 + nn;
    const float v = tanhf(fmaf(tile[b][nn], w_in[n], b0[n]));
    sT[(n << 6) + b] = v;
  }
}

// One 64-lane team per node: each lane owns one batch element b.
// Gather of row src reads 64 contiguous floats (256B) -> perfectly coalesced.
__global__ __launch_bounds__(256) void fuse_step_kernel(
    const float* __restrict__ sPrev,   // [N][64] transposed state
    const float* __restrict__ gw,      // [N][D] this step's edge weights
    const int*   __restrict__ gsrc,    // [N][D] this step's edge sources
    const float* __restrict__ bias,    // [N]
    float* __restrict__ dst,
    int finalLayout)                   // 0: [N][64] transposed, 1: [B][N] final
{
  __shared__ int   s_src[NODES_PER_BLOCK * DEG];
  __shared__ float s_w  [NODES_PER_BLOCK * DEG];
  __shared__ float s_out[NODES_PER_BLOCK][BATCH];   // final-layout store staging

  const int node0 = blockIdx.x * NODES_PER_BLOCK;
  const int e0    = node0 * DEG;

#if USE_TDM
  // Only wave 0 issues the TDM ops (EXEC is ignored by tensor instructions),
  // waits on TENSORcnt, then the whole block syncs.
  if (threadIdx.x == 0) {
    tdm_load_row_to_lds((uint32_t)(uintptr_t)&s_src[0], gsrc + e0, NODES_PER_BLOCK * DEG);
    tdm_load_row_to_lds((uint32_t)(uintptr_t)&s_w[0],   gw   + e0, NODES_PER_BLOCK * DEG);
    __builtin_prefetch(gw + e0 + NODES_PER_BLOCK * DEG, 0, 0);   // next block's rows
    __builtin_amdgcn_s_wait_tensorcnt(0);
  }
#else
  if (threadIdx.x < NODES_PER_BLOCK * DEG) {
    s_src[threadIdx.x] = gsrc[e0 + threadIdx.x];
  } else if (threadIdx.x < 2 * NODES_PER_BLOCK * DEG) {
    const int i = threadIdx.x - NODES_PER_BLOCK * DEG;
    s_w[i] = gw[e0 + i];
  }
#endif
  __syncthreads();

  const int team = threadIdx.x >> 6;          // node within block
  const int b    = threadIdx.x & (BATCH - 1); // batch lane
  const int n    = node0 + team;

  const int*   sp = s_src + team * DEG;
  const float* wp = s_w   + team * DEG;

  // Two accumulator chains: halves the serial FMA dependency latency.
  float acc0 = 0.0f, acc1 = 0.0f;
#pragma unroll
  for (int d = 0; d < DEG; d += 2) {
    const float* r0 = sPrev + ((uint32_t)sp[d]     << 6);  // 256B contiguous segment
    const float* r1 = sPrev + ((uint32_t)sp[d + 1] << 6);
    acc0 = fmaf(wp[d],     r0[b], acc0);
    acc1 = fmaf(wp[d + 1], r1[b], acc1);
  }

  const float v = tanhf((acc0 + acc1) + bias[n]);

  if (finalLayout) {
    // Stage through LDS so each batch row writes one 16B b128 store
    // (4 contiguous node outputs) instead of 4 scattered 4B stores.
    s_out[team][b] = v;
    __syncthreads();
    if (threadIdx.x < BATCH) {
      const int bb = threadIdx.x;
      v4f o;
      o.x = s_out[0][bb];
      o.y = s_out[1][bb];
      o.z = s_out[2][bb];
      o.w = s_out[3][bb];
      // Output is write-once with no reuse: non-temporal store.
      __builtin_nontemporal_store(o, (v4f*)(dst + (uint32_t)bb * N_NODES + node0));
    }
  } else {
    dst[((uint32_t)n << 6) + b] = v;   // transposed state, coalesced
  }
}

extern "C" void kernel_launch(void* const* d_in, const int* in_sizes, int n_in,
                              void* d_out, int out_size, void* d_ws, size_t ws_size,
                              hipStream_t stream) {
  (void)in_sizes; (void)n_in; (void)out_size; (void)ws_size;

  const float* x        = (const float*)d_in[0];
  const float* w_in     = (const float*)d_in[1];
  const float* edge_w   = (const float*)d_in[2];
  const float* biases   = (const float*)d_in[3];
  const int*   edge_src = (const int*)d_in[4];

  float* outF = (float*)d_out;   // doubles as transposed scratch until last step
  float* wsF  = (float*)d_ws;    // needs N*B*4 = 4 MB

  // state s0^T -> d_out(scratch)
  init_state_kernel<<<N_NODES / TILE_N, 256, 0, stream>>>(x, w_in, biases, outF);

  // Ping-pong: even step reads out-scratch -> ws, odd reads ws -> out-scratch.
  // Step 5 (last) reads ws and writes the final [B,N] layout into d_out.
  for (int s = 0; s < NSTEPS; ++s) {
    const float* src = (s % 2 == 0) ? outF : wsF;
    float*       dst = (s % 2 == 0) ? wsF  : outF;
    const int finalLayout = (s == NSTEPS - 1) ? 1 : 0;
    fuse_step_kernel<<<N_NODES / NODES_PER_BLOCK, 256, 0, stream>>>(
        src,
        edge_w   + (size_t)s * N_NODES * DEG,
        edge_src + (size_t)s * N_NODES * DEG,
        biases   + (size_t)(s + 1) * N_NODES,
        dst, finalLayout);
  }
}